// GATLayer_1151051235751
// MI455X (gfx1250) — compile-verified
//
#include <hip/hip_runtime.h>
#include <math.h>

#define N_NODES 50000
#define F       256          // F_IN == F_OUT == 256
#define EDIM    64
#define LRELU_ALPHA 0.2f

typedef __attribute__((ext_vector_type(2))) float v2f;
typedef __attribute__((ext_vector_type(8))) float v8f;

// ---------------------------------------------------------------------------
// float atomic max via integer punning (works for any sign mix when the cell
// is initialized to -inf).
__device__ __forceinline__ void atomicMaxF(float* addr, float val) {
    if (val >= 0.0f) {
        atomicMax((int*)addr, __float_as_int(val));
    } else {
        atomicMin((unsigned int*)addr, __float_as_uint(val));
    }
}

// ---------------------------------------------------------------------------
// K0: init output (zeros), segment max (-inf), denom (zeros)
__global__ void init_kernel(float* __restrict__ out,
                            float* __restrict__ mval,
                            float* __restrict__ denom) {
    size_t i = (size_t)blockIdx.x * blockDim.x + threadIdx.x;
    if (i < (size_t)N_NODES * F) out[i] = 0.0f;
    if (i < (size_t)N_NODES) {
        mval[i]  = -INFINITY;
        denom[i] = 0.0f;
    }
}

// ---------------------------------------------------------------------------
// K1: z = h @ W^T using V_WMMA_F32_16X16X4_F32.
// Block = 128 threads = 4 waves. Block computes a 16x256 tile of z; each wave
// computes 16x64 via 4 accumulators. K looped in steps of 4.
// No divergence: EXEC is all-ones throughout (WMMA requirement).
__global__ __launch_bounds__(128) void gemm_z(const float* __restrict__ h,
                                              const float* __restrict__ W,
                                              float* __restrict__ z) {
    const int wave   = threadIdx.x >> 5;
    const int lane   = threadIdx.x & 31;
    const int lanelo = lane & 15;
    const int hi     = lane >> 4;            // 0 or 1
    const int m0     = blockIdx.x * 16;      // 50000 % 16 == 0, no tail
    const int n0     = wave * 64;

    v8f acc0 = {}, acc1 = {}, acc2 = {}, acc3 = {};

    // A fragment row (16x4, f32): lane L<16 -> row m0+L holds K=0,1;
    // lane L>=16 -> same row holds K=2,3.  => float2 load at col k + 2*hi.
    const float* arow = h + (size_t)(m0 + lanelo) * F;
    // B fragment (4x16): B[k'][n] = W[n][k+k'] (z = h @ W^T). Same striping.
    const float* brow0 = W + (size_t)(n0 +  0 + lanelo) * F;
    const float* brow1 = W + (size_t)(n0 + 16 + lanelo) * F;
    const float* brow2 = W + (size_t)(n0 + 32 + lanelo) * F;
    const float* brow3 = W + (size_t)(n0 + 48 + lanelo) * F;

    for (int k = 0; k < F; k += 4) {
        const int ko = k + 2 * hi;
        v2f a  = *(const v2f*)(arow  + ko);
        v2f b0 = *(const v2f*)(brow0 + ko);
        v2f b1 = *(const v2f*)(brow1 + ko);
        v2f b2 = *(const v2f*)(brow2 + ko);
        v2f b3 = *(const v2f*)(brow3 + ko);
        acc0 = __builtin_amdgcn_wmma_f32_16x16x4_f32(false, a, false, b0,
                                                     (short)0, acc0, false, false);
        acc1 = __builtin_amdgcn_wmma_f32_16x16x4_f32(false, a, false, b1,
                                                     (short)0, acc1, false, false);
        acc2 = __builtin_amdgcn_wmma_f32_16x16x4_f32(false, a, false, b2,
                                                     (short)0, acc2, false, false);
        acc3 = __builtin_amdgcn_wmma_f32_16x16x4_f32(false, a, false, b3,
                                                     (short)0, acc3, false, false);
    }

    // C/D layout: VGPR r -> (M=r, lanes 0-15) and (M=r+8, lanes 16-31), N=lanelo.
#pragma unroll
    for (int r = 0; r < 8; ++r) {
        const size_t m = (size_t)(m0 + r + 8 * hi);
        z[m * F + n0 +  0 + lanelo] = acc0[r];
        z[m * F + n0 + 16 + lanelo] = acc1[r];
        z[m * F + n0 + 32 + lanelo] = acc2[r];
        z[m * F + n0 + 48 + lanelo] = acc3[r];
    }
}

// ---------------------------------------------------------------------------
// K2: per-node attention pre-dots: s1[n] = z[n]·a_src, s2[n] = z[n]·a_dst.
// One wave per node.
__global__ __launch_bounds__(256) void node_dots(const float* __restrict__ z,
                                                 const float* __restrict__ A,
                                                 float* __restrict__ s1,
                                                 float* __restrict__ s2) {
    const int node = blockIdx.x * 8 + (threadIdx.x >> 5);
    if (node >= N_NODES) return;
    const int lane = threadIdx.x & 31;
    const float* zr = z + (size_t)node * F;
    float p1 = 0.0f, p2 = 0.0f;
#pragma unroll
    for (int c = lane; c < F; c += 32) {
        const float v = zr[c];
        p1 += v * A[c];         // a_src = A[0:256]
        p2 += v * A[F + c];     // a_dst = A[256:512]
    }
    for (int off = 16; off > 0; off >>= 1) {
        p1 += __shfl_xor(p1, off, 32);
        p2 += __shfl_xor(p2, off, 32);
    }
    if (lane == 0) { s1[node] = p1; s2[node] = p2; }
}

// ---------------------------------------------------------------------------
// K3: per-edge logits, leaky-relu, segment max (atomic). One wave per edge:
// EDIM=64 -> float2 per lane, butterfly reduce.
__global__ __launch_bounds__(256) void edge_logits(const float* __restrict__ e_feat,
                                                   const float* __restrict__ A,
                                                   const int* __restrict__ src,
                                                   const int* __restrict__ dst,
                                                   const float* __restrict__ s1,
                                                   const float* __restrict__ s2,
                                                   float* __restrict__ e_arr,
                                                   float* __restrict__ mval,
                                                   int E) {
    const int edge = blockIdx.x * 8 + (threadIdx.x >> 5);
    if (edge >= E) return;
    const int lane = threadIdx.x & 31;
    v2f ef = *(const v2f*)(e_feat + (size_t)edge * EDIM + 2 * lane);
    v2f ae = *(const v2f*)(A + 2 * F + 2 * lane);     // a_e = A[512:576]
    float p = ef[0] * ae[0] + ef[1] * ae[1];
    for (int off = 16; off > 0; off >>= 1) p += __shfl_xor(p, off, 32);
    if (lane == 0) {
        const int sN = src[edge], dN = dst[edge];
        const float logit = p + s1[sN] + s2[dN];
        const float e = (logit > 0.0f) ? logit : LRELU_ALPHA * logit;
        e_arr[edge] = e;
        atomicMaxF(&mval[dN], e);
    }
}

// ---------------------------------------------------------------------------
// K4: ex = exp(e - m[dst]); denom[dst] += ex. One thread per edge.
__global__ void edge_exp(const float* __restrict__ e_arr,
                         const int* __restrict__ dst,
                         const float* __restrict__ mval,
                         float* __restrict__ ex_arr,
                         float* __restrict__ denom,
                         int E) {
    const int i = blockIdx.x * blockDim.x + threadIdx.x;
    if (i >= E) return;
    const int d = dst[i];
    const float ex = expf(e_arr[i] - mval[d]);
    ex_arr[i] = ex;
    atomicAdd(denom + d, ex);
}

// ---------------------------------------------------------------------------
// K5: h_out[dst] += (ex/denom[dst]) * z[src]. One wave per edge, lanes striped
// across the 256 columns: coalesced loads + coalesced global_atomic_add_f32.
// This is the bandwidth bottleneck (~1.6 GB of traffic).
__global__ __launch_bounds__(256) void scatter_out(const float* __restrict__ z,
                                                   const int* __restrict__ src,
                                                   const int* __restrict__ dst,
                                                   const float* __restrict__ ex_arr,
                                                   const float* __restrict__ denom,
                                                   float* __restrict__ out,
                                                   int E) {
    const int edge = blockIdx.x * 8 + (threadIdx.x >> 5);
    if (edge >= E) return;
    const int lane = threadIdx.x & 31;
    const int sN = src[edge], dN = dst[edge];
    const float alpha = ex_arr[edge] / denom[dN];
    const float* zr = z + (size_t)sN * F;
    float* orow = out + (size_t)dN * F;
#pragma unroll
    for (int c = lane; c < F; c += 32) {
        atomicAdd(orow + c, alpha * zr[c]);
    }
}

// ---------------------------------------------------------------------------
extern "C" void kernel_launch(void* const* d_in, const int* in_sizes, int n_in,
                              void* d_out, int out_size, void* d_ws, size_t ws_size,
                              hipStream_t stream) {
    const float* h      = (const float*)d_in[0];
    const float* e_feat = (const float*)d_in[1];
    const float* W      = (const float*)d_in[2];
    const float* A      = (const float*)d_in[3];
    const int*   src    = (const int*)d_in[4];
    const int*   dst    = (const int*)d_in[5];
    float*       out    = (float*)d_out;
    const int E = in_sizes[4];

    // Workspace layout (f32): z | s1 | s2 | m | denom | e | ex  (~58 MB)
    float* ws    = (float*)d_ws;
    float* z     = ws;
    float* s1    = z  + (size_t)N_NODES * F;
    float* s2    = s1 + N_NODES;
    float* mval  = s2 + N_NODES;
    float* denom = mval + N_NODES;
    float* e_arr = denom + N_NODES;
    float* ex_arr= e_arr + E;

    // K0: init
    {
        const size_t total = (size_t)N_NODES * F;
        init_kernel<<<(unsigned)((total + 255) / 256), 256, 0, stream>>>(out, mval, denom);
    }
    // K1: z = h @ W^T  (WMMA f32)
    gemm_z<<<N_NODES / 16, 128, 0, stream>>>(h, W, z);
    // K2: per-node dots
    node_dots<<<(N_NODES + 7) / 8, 256, 0, stream>>>(z, A, s1, s2);
    // K3: edge logits + leaky relu + segment max
    edge_logits<<<(E + 7) / 8, 256, 0, stream>>>(e_feat, A, src, dst, s1, s2,
                                                 e_arr, mval, E);
    // K4: exp + denom
    edge_exp<<<(E + 255) / 256, 256, 0, stream>>>(e_arr, dst, mval, ex_arr, denom, E);
    // K5: weighted scatter-add
    scatter_out<<<(E + 7) / 8, 256, 0, stream>>>(z, src, dst, ex_arr, denom, out, E);
}